// POSExtractor_14336600834464
// MI455X (gfx1250) — compile-verified
//
#include <hip/hip_runtime.h>
#include <math.h>

// POS rPPG extractor, restructured algebraically:
//   per-window closed-form coefficients (9 sliding sums) ->
//   H[b,n] = x0[n]*SS(c0) + x1[n]*SS(c1) + x2[n]*SS(c2) - SS(cm)
// where SS is a 48-tap ones-convolution done with V_WMMA_F32_16X16X4_F32.
// Coefficient arrays are stored with a 64-float zero pad in front so the
// WMMA B-operand loads need no bounds checks (no exec-mask churn).

#define B_ 32
#define N_ 8192
#define L_ 48
#define W_ (N_ - L_) /* 8144 */
#define PAD_ 64
#define PADN_ (N_ + PAD_) /* per-array stride, zero pad in front */

typedef __attribute__((ext_vector_type(2))) float v2f;
typedef __attribute__((ext_vector_type(8))) float v8f;

// ---------------------------------------------------------------------------
// Kernel 1: per-window coefficients c0,c1,c2,cm.
// Writes zeros to the 64-float front pad and for w in [W, N) (tail), so the
// convolution kernel can read unconditionally.
// block: 256 threads = 256 windows of one batch; x staged through LDS.
// ---------------------------------------------------------------------------
__global__ __launch_bounds__(256) void pos_window_coeffs(
    const float* __restrict__ x, float* __restrict__ c) {
  const int b  = blockIdx.y;
  const int w0 = blockIdx.x * 256;
  const int t  = threadIdx.x;

  __shared__ float sx0[304], sx1[304], sx2[304];
  for (int i = t; i < 304; i += 256) {
    int n = w0 + i;
    float a0 = 0.f, a1 = 0.f, a2 = 0.f;
    if (n < N_) {
      const float* p = x + ((size_t)b * N_ + n) * 3;
      a0 = p[0]; a1 = p[1]; a2 = p[2];
    }
    sx0[i] = a0; sx1[i] = a1; sx2[i] = a2;
  }
  __syncthreads();

  // front pad: first block of each batch zeroes c[a][b][0..63]
  if (blockIdx.x == 0 && t < PAD_) {
    const size_t pbase = (size_t)b * PADN_ + t;
    c[0*(size_t)B_*PADN_ + pbase] = 0.f;
    c[1*(size_t)B_*PADN_ + pbase] = 0.f;
    c[2*(size_t)B_*PADN_ + pbase] = 0.f;
    c[3*(size_t)B_*PADN_ + pbase] = 0.f;
  }

  const int w = w0 + t;
  float c0 = 0.f, c1 = 0.f, c2 = 0.f, cm = 0.f;
  if (w < W_) {
    float s0=0,s1=0,s2=0,q00=0,q11=0,q22=0,q01=0,q02=0,q12=0;
    #pragma unroll 8
    for (int l = 0; l < L_; ++l) {
      float a0 = sx0[t+l], a1 = sx1[t+l], a2 = sx2[t+l];
      s0 += a0; s1 += a1; s2 += a2;
      q00 += a0*a0; q11 += a1*a1; q22 += a2*a2;
      q01 += a0*a1; q02 += a0*a2; q12 += a1*a2;
    }
    const float m0 = s0*(1.0f/L_), m1 = s1*(1.0f/L_), m2 = s2*(1.0f/L_);
    const float i0 = 1.0f/m0, i1 = 1.0f/m1, i2 = 1.0f/m2;
    // sum(S0^2), sum(S1^2) in closed form (mean(S0)=mean(S1)=0 analytically)
    const float ss0 = q11*i1*i1 - 2.0f*q12*i1*i2 + q22*i2*i2;
    const float ss1 = 4.0f*q00*i0*i0 + q11*i1*i1 + q22*i2*i2
                    - 4.0f*q01*i0*i1 - 4.0f*q02*i0*i2 + 2.0f*q12*i1*i2;
    const float r = sqrtf(ss0 / ss1);  // std0/std1 (ddof cancels in ratio)
    // tiny numerical residue of mean(h) kept for fidelity
    const float sum0 = s1*i1 - s2*i2;
    const float sum1 = -2.0f*s0*i0 + s1*i1 + s2*i2;
    cm = (sum0 + r*sum1) * (1.0f/L_);
    c0 = -2.0f*r*i0;
    c1 = (1.0f + r)*i1;
    c2 = (r - 1.0f)*i2;
  }
  const size_t base = (size_t)b * PADN_ + PAD_ + w;
  c[0*(size_t)B_*PADN_ + base] = c0;
  c[1*(size_t)B_*PADN_ + base] = c1;
  c[2*(size_t)B_*PADN_ + base] = c2;
  c[3*(size_t)B_*PADN_ + base] = cm;
}

// ---------------------------------------------------------------------------
// Kernel 2: 48-tap ones-convolution over window index via f32 WMMA,
// then elementwise combine with x -> H.
// One wave: 16 output positions (M) x 16 streams (N = 4 arrays x 4 batches),
// K = 64 input positions accumulated as 16 x V_WMMA_F32_16X16X4_F32.
// All B-operand loads unconditional thanks to the front zero pad.
// ---------------------------------------------------------------------------
__global__ __launch_bounds__(256) void pos_overlap_add(
    const float* __restrict__ x, const float* __restrict__ c,
    float* __restrict__ out) {
  const int wave = threadIdx.x >> 5;
  const int lane = threadIdx.x & 31;
  const int g    = blockIdx.y;                 // batch group (4 batches)
  const int tb   = blockIdx.x * 8 + wave;      // 16-wide output block
  const int T0   = tb * 16;

  const int khalf = lane >> 4;                 // K-half select (ISA A/B layout)
  const int scol  = lane & 15;                 // M row for A, N col for B
  const int a_idx = scol & 3;                  // which coefficient array
  const int bb    = g * 4 + (scol >> 2);       // which batch

  // padded base: element j of the K=64 slab lives at [PAD_ + T0 - 48 + j]
  const float* cptr = c + ((size_t)a_idx * B_ + bb) * PADN_
                        + (PAD_ - 48) + T0 + 2 * khalf;

  v8f acc = {0.f,0.f,0.f,0.f,0.f,0.f,0.f,0.f};
  #pragma unroll
  for (int kb = 0; kb < 16; ++kb) {
    const int j0 = 4*kb + 2*khalf;             // this lane's K indices j0, j0+1
    const v2f bv = *(const v2f*)(cptr + 4*kb); // always in-range (front pad)
    // Banded Toeplitz: T[m][j] = 1 iff m+1 <= j <= m+48
    v2f av;
    av.x = (j0     >= scol + 1 && j0     <= scol + 48) ? 1.0f : 0.0f;
    av.y = (j0 + 1 >= scol + 1 && j0 + 1 <= scol + 48) ? 1.0f : 0.0f;
    acc = __builtin_amdgcn_wmma_f32_16x16x4_f32(
        /*neg_a=*/false, av, /*neg_b=*/false, bv,
        /*c_mod=*/(short)0, acc, /*reuse_a=*/false, /*reuse_b=*/false);
  }

  // D layout: VGPR v -> row v (lanes 0-15) / row v+8 (lanes 16-31), col = scol
  __shared__ float smem[8][256];
  #pragma unroll
  for (int v = 0; v < 8; ++v)
    smem[wave][scol * 16 + (v + 8 * khalf)] = acc[v];
  __syncthreads();

  // combine: 64 outputs per wave (4 batches x 16 positions)
  #pragma unroll
  for (int rep = 0; rep < 2; ++rep) {
    const int o  = lane + rep * 32;
    const int qi = o >> 4;                     // batch within group
    const int i  = o & 15;                     // position within block
    const int n  = T0 + i;
    const int b  = g * 4 + qi;
    const float V0 = smem[wave][(qi*4 + 0)*16 + i];
    const float V1 = smem[wave][(qi*4 + 1)*16 + i];
    const float V2 = smem[wave][(qi*4 + 2)*16 + i];
    const float V3 = smem[wave][(qi*4 + 3)*16 + i];
    const float* xp = x + ((size_t)b * N_ + n) * 3;
    out[(size_t)b * N_ + n] = xp[0]*V0 + xp[1]*V1 + xp[2]*V2 - V3;
  }
}

// ---------------------------------------------------------------------------
extern "C" void kernel_launch(void* const* d_in, const int* in_sizes, int n_in,
                              void* d_out, int out_size, void* d_ws, size_t ws_size,
                              hipStream_t stream) {
  const float* x = (const float*)d_in[0];
  float* cws = (float*)d_ws;                   // 4 * B * (N+64) floats ~ 4.1 MB
  float* out = (float*)d_out;

  dim3 gA(N_ / 256, B_);                       // 32 x 32 blocks
  pos_window_coeffs<<<gA, 256, 0, stream>>>(x, cws);

  dim3 gB((N_ / 16) / 8, B_ / 4);              // 64 x 8 blocks, 8 waves each
  pos_overlap_add<<<gB, 256, 0, stream>>>(x, cws, out);
}